// Transformer_45406394254036
// MI455X (gfx1250) — compile-verified
//
#include <hip/hip_runtime.h>
#include <hip/hip_bf16.h>

// ---------------------------------------------------------------------------
// CDNA5 (gfx1250, wave32) implementation of the reference transformer.
// All dense linear algebra runs on the WMMA pipe via V_WMMA_F32_16X16X4_F32.
// ---------------------------------------------------------------------------

typedef __attribute__((ext_vector_type(2))) float v2f;
typedef __attribute__((ext_vector_type(8))) float v8f;

__device__ __forceinline__ v8f wmma_f32(v2f a, v2f b, v8f c) {
  // 8-arg form: (neg_a, A, neg_b, B, c_mod, C, reuse_a, reuse_b)
  return __builtin_amdgcn_wmma_f32_16x16x4_f32(false, a, false, b, (short)0, c,
                                               false, false);
}

#define NEGV (-65504.0f)

// ---------------------------------------------------------------------------
// GEMM (small-M): C[M,N] = A[M,K] @ W[K,N] + bias (optional ReLU).
// Block = 256 threads = 8 waves; wave tile 16(M)x64(N); block tile 64x128.
// Requires M%64==0, N%128==0, K%4==0.
// ---------------------------------------------------------------------------
__global__ void gemm_kernel(const float* __restrict__ A, int lda,
                            const float* __restrict__ W, int ldw,
                            const float* __restrict__ bias,
                            float* __restrict__ C, int ldc,
                            int K, int relu) {
  const int tid  = threadIdx.x;
  const int lane = tid & 31;
  const int wave = tid >> 5;
  const int half = lane >> 4;   // 0: lanes 0-15, 1: lanes 16-31
  const int l16  = lane & 15;
  const int mw = wave & 3, nw = wave >> 2;
  const size_t mbase = (size_t)blockIdx.y * 64 + mw * 16;
  const int    nbase = blockIdx.x * 128 + nw * 64;

  v8f acc0 = {}, acc1 = {}, acc2 = {}, acc3 = {};
  const float* arow = A + (mbase + l16) * (size_t)lda;

  for (int kk = 0; kk < K; kk += 4) {
    const float* ap = arow + kk + half * 2;
    v2f a; a[0] = ap[0]; a[1] = ap[1];
    const float* wb = W + (size_t)(kk + half * 2) * ldw + nbase + l16;
    v2f b0; b0[0] = wb[0];  b0[1] = wb[ldw];
    v2f b1; b1[0] = wb[16]; b1[1] = wb[ldw + 16];
    v2f b2; b2[0] = wb[32]; b2[1] = wb[ldw + 32];
    v2f b3; b3[0] = wb[48]; b3[1] = wb[ldw + 48];
    acc0 = wmma_f32(a, b0, acc0);
    acc1 = wmma_f32(a, b1, acc1);
    acc2 = wmma_f32(a, b2, acc2);
    acc3 = wmma_f32(a, b3, acc3);
  }

#pragma unroll
  for (int r = 0; r < 8; ++r) {
    const size_t row = mbase + r + half * 8;
    float* crow = C + row * (size_t)ldc;
    int c0 = nbase + l16;
    float v0 = acc0[r] + bias[c0];
    float v1 = acc1[r] + bias[c0 + 16];
    float v2 = acc2[r] + bias[c0 + 32];
    float v3 = acc3[r] + bias[c0 + 48];
    if (relu) {
      v0 = fmaxf(v0, 0.f); v1 = fmaxf(v1, 0.f);
      v2 = fmaxf(v2, 0.f); v3 = fmaxf(v3, 0.f);
    }
    crow[c0] = v0; crow[c0 + 16] = v1; crow[c0 + 32] = v2; crow[c0 + 48] = v3;
  }
}

// ---------------------------------------------------------------------------
// GEMM (large-M): wave tile 32(M)x64(N) -> 8 WMMA per K-step amortize the
// B-fragment loads over two M-subtiles (halves VMEM dwords per FLOP vs the
// small kernel).  Block = 8 waves = 4 M-pos x 2 N-pos; block tile 128x128.
// Requires M%128==0, N%128==0, K%4==0.
// ---------------------------------------------------------------------------
__global__ void gemm_big_kernel(const float* __restrict__ A, int lda,
                                const float* __restrict__ W, int ldw,
                                const float* __restrict__ bias,
                                float* __restrict__ C, int ldc,
                                int K, int relu) {
  const int tid  = threadIdx.x;
  const int lane = tid & 31;
  const int wave = tid >> 5;
  const int half = lane >> 4;
  const int l16  = lane & 15;
  const int mw = wave & 3, nw = wave >> 2;
  const size_t mbase = (size_t)blockIdx.y * 128 + mw * 32;
  const int    nbase = blockIdx.x * 128 + nw * 64;

  v8f acc00 = {}, acc01 = {}, acc02 = {}, acc03 = {};
  v8f acc10 = {}, acc11 = {}, acc12 = {}, acc13 = {};
  const float* arow0 = A + (mbase + l16) * (size_t)lda;
  const float* arow1 = A + (mbase + 16 + l16) * (size_t)lda;

  for (int kk = 0; kk < K; kk += 4) {
    const float* ap0 = arow0 + kk + half * 2;
    const float* ap1 = arow1 + kk + half * 2;
    v2f a0; a0[0] = ap0[0]; a0[1] = ap0[1];
    v2f a1; a1[0] = ap1[0]; a1[1] = ap1[1];
    const float* wb = W + (size_t)(kk + half * 2) * ldw + nbase + l16;
    v2f b0; b0[0] = wb[0];  b0[1] = wb[ldw];
    v2f b1; b1[0] = wb[16]; b1[1] = wb[ldw + 16];
    v2f b2; b2[0] = wb[32]; b2[1] = wb[ldw + 32];
    v2f b3; b3[0] = wb[48]; b3[1] = wb[ldw + 48];
    acc00 = wmma_f32(a0, b0, acc00);
    acc01 = wmma_f32(a0, b1, acc01);
    acc02 = wmma_f32(a0, b2, acc02);
    acc03 = wmma_f32(a0, b3, acc03);
    acc10 = wmma_f32(a1, b0, acc10);
    acc11 = wmma_f32(a1, b1, acc11);
    acc12 = wmma_f32(a1, b2, acc12);
    acc13 = wmma_f32(a1, b3, acc13);
  }

  const int c0 = nbase + l16;
#pragma unroll
  for (int r = 0; r < 8; ++r) {
    const size_t row0 = mbase + r + half * 8;
    const size_t row1 = row0 + 16;
    float* cr0 = C + row0 * (size_t)ldc;
    float* cr1 = C + row1 * (size_t)ldc;
    float u0 = acc00[r] + bias[c0];
    float u1 = acc01[r] + bias[c0 + 16];
    float u2 = acc02[r] + bias[c0 + 32];
    float u3 = acc03[r] + bias[c0 + 48];
    float w0 = acc10[r] + bias[c0];
    float w1 = acc11[r] + bias[c0 + 16];
    float w2 = acc12[r] + bias[c0 + 32];
    float w3 = acc13[r] + bias[c0 + 48];
    if (relu) {
      u0 = fmaxf(u0, 0.f); u1 = fmaxf(u1, 0.f);
      u2 = fmaxf(u2, 0.f); u3 = fmaxf(u3, 0.f);
      w0 = fmaxf(w0, 0.f); w1 = fmaxf(w1, 0.f);
      w2 = fmaxf(w2, 0.f); w3 = fmaxf(w3, 0.f);
    }
    cr0[c0] = u0; cr0[c0 + 16] = u1; cr0[c0 + 32] = u2; cr0[c0 + 48] = u3;
    cr1[c0] = w0; cr1[c0 + 16] = w1; cr1[c0 + 32] = w2; cr1[c0 + 48] = w3;
  }
}

// ---------------------------------------------------------------------------
// Attention: O = softmax(mask(Q K^T * scale)) V  for one (batch, head, qtile).
// grid = (Lq/16, heads, B). Q/K/V/O are (B, L, ld) with head columns h*D..
// Phase 1: WMMA scores -> LDS.  Phase 2: masked softmax (shuffle reduce).
// Phase 3: WMMA P*V from LDS.  Lk <= 256, D%16==0.
// ---------------------------------------------------------------------------
#define LKMAX 256
__global__ void attn_kernel(const float* __restrict__ Q,
                            const float* __restrict__ K,
                            const float* __restrict__ V,
                            float* __restrict__ O,
                            const unsigned char* __restrict__ mask,
                            int Lq, int Lk, int D, int ld, float scale) {
  __shared__ float S[16][LKMAX + 4];
  const int tid  = threadIdx.x;
  const int lane = tid & 31;
  const int wave = tid >> 5;
  const int half = lane >> 4;
  const int l16  = lane & 15;
  const int qt = blockIdx.x;
  const int h  = blockIdx.y;
  const int b  = blockIdx.z;
  const size_t qrow0 = (size_t)b * Lq + qt * 16;
  const size_t krow0 = (size_t)b * Lk;
  const int coff = h * D;

  // ---- Phase 1: S = scale * Q K^T (masked) ----
  const int nkt = Lk >> 4;
  for (int kt = wave; kt < nkt; kt += 8) {
    v8f acc = {};
    for (int d = 0; d < D; d += 4) {
      const float* ap = Q + (qrow0 + l16) * (size_t)ld + coff + d + half * 2;
      v2f a; a[0] = ap[0]; a[1] = ap[1];
      const float* bp = K + (krow0 + kt * 16 + l16) * (size_t)ld + coff + d + half * 2;
      v2f bb; bb[0] = bp[0]; bb[1] = bp[1];
      acc = wmma_f32(a, bb, acc);
    }
    const int kpos = kt * 16 + l16;
    const bool mk = (mask != nullptr) && (mask[krow0 + kpos] != 0);
#pragma unroll
    for (int r = 0; r < 8; ++r) {
      const int m = r + half * 8;
      S[m][kpos] = mk ? NEGV : acc[r] * scale;
    }
  }
  __syncthreads();

  // ---- Phase 2: row softmax (16 threads per row) ----
  {
    const int row = tid >> 4;
    const int idx = tid & 15;
    float mx = -3.0e38f;
    for (int j = idx; j < Lk; j += 16) mx = fmaxf(mx, S[row][j]);
#pragma unroll
    for (int o = 8; o > 0; o >>= 1) mx = fmaxf(mx, __shfl_xor(mx, o, 32));
    float sum = 0.f;
    for (int j = idx; j < Lk; j += 16) {
      float e = expf(S[row][j] - mx);
      S[row][j] = e;
      sum += e;
    }
#pragma unroll
    for (int o = 8; o > 0; o >>= 1) sum += __shfl_xor(sum, o, 32);
    const float inv = 1.0f / sum;
    for (int j = idx; j < Lk; j += 16) S[row][j] *= inv;
  }
  __syncthreads();

  // ---- Phase 3: O = P V ----
  const int nnt = D >> 4;
  for (int nt = wave; nt < nnt; nt += 8) {
    v8f acc = {};
    for (int kk = 0; kk < Lk; kk += 4) {
      v2f a;
      a[0] = S[l16][kk + half * 2];
      a[1] = S[l16][kk + half * 2 + 1];
      const float* bp = V + (krow0 + kk + half * 2) * (size_t)ld + coff + nt * 16 + l16;
      v2f bb; bb[0] = bp[0]; bb[1] = bp[ld];
      acc = wmma_f32(a, bb, acc);
    }
#pragma unroll
    for (int r = 0; r < 8; ++r) {
      const int m = r + half * 8;
      O[(qrow0 + m) * (size_t)ld + coff + nt * 16 + l16] = acc[r];
    }
  }
}

// ---------------------------------------------------------------------------
// GEMV for N==1 heads: out[m] = A[m,:].w + b[0].  One wave per row.
// ---------------------------------------------------------------------------
__global__ void gemv1_kernel(const float* __restrict__ A, int lda,
                             const float* __restrict__ w,
                             const float* __restrict__ b,
                             float* __restrict__ out, int K) {
  const int lane = threadIdx.x & 31;
  const int wave = threadIdx.x >> 5;
  const size_t row = (size_t)blockIdx.x * 8 + wave;
  const float* ar = A + row * (size_t)lda;
  float s = 0.f;
  for (int k = lane; k < K; k += 32) s += ar[k] * w[k];
#pragma unroll
  for (int o = 16; o > 0; o >>= 1) s += __shfl_xor(s, o, 32);
  if (lane == 0) out[row] = s + b[0];
}

// ---------------------------------------------------------------------------
// LayerNorm with optional residual: out = g*norm(x+y) + b. One block per row.
// ---------------------------------------------------------------------------
__global__ void ln_kernel(float* __restrict__ out,
                          const float* __restrict__ x,
                          const float* __restrict__ y,
                          const float* __restrict__ g,
                          const float* __restrict__ bb, int D) {
  __shared__ float red[256];
  const size_t row = blockIdx.x;
  const float* xr = x + row * (size_t)D;
  const float* yr = y ? (y + row * (size_t)D) : nullptr;
  float s1 = 0.f, s2 = 0.f;
  for (int d = threadIdx.x; d < D; d += 256) {
    float v = xr[d] + (yr ? yr[d] : 0.f);
    s1 += v; s2 += v * v;
  }
  red[threadIdx.x] = s1; __syncthreads();
  for (int o = 128; o > 0; o >>= 1) {
    if (threadIdx.x < o) red[threadIdx.x] += red[threadIdx.x + o];
    __syncthreads();
  }
  const float mean = red[0] / D;
  __syncthreads();
  red[threadIdx.x] = s2; __syncthreads();
  for (int o = 128; o > 0; o >>= 1) {
    if (threadIdx.x < o) red[threadIdx.x] += red[threadIdx.x + o];
    __syncthreads();
  }
  const float var = red[0] / D - mean * mean;
  const float inv = rsqrtf(var + 1e-5f);
  for (int d = threadIdx.x; d < D; d += 256) {
    float v = xr[d] + (yr ? yr[d] : 0.f);
    out[row * (size_t)D + d] = g[d] * (v - mean) * inv + bb[d];
  }
}

// ---------------------------------------------------------------------------
// Masked softmax over sequence dim: att[b, :] = softmax(where(mask, NEG, x)).
// One block per batch; L <= 256.
// ---------------------------------------------------------------------------
__global__ void softmax_mask_kernel(float* __restrict__ att,
                                    const float* __restrict__ logits,
                                    const unsigned char* __restrict__ mask,
                                    int L) {
  __shared__ float red[256];
  const int b = blockIdx.x, t = threadIdx.x;
  float x = -3.0e38f;
  if (t < L) x = mask[(size_t)b * L + t] ? NEGV : logits[(size_t)b * L + t];
  red[t] = x; __syncthreads();
  for (int o = 128; o > 0; o >>= 1) {
    if (t < o) red[t] = fmaxf(red[t], red[t + o]);
    __syncthreads();
  }
  const float mx = red[0];
  __syncthreads();
  const float e = (t < L) ? expf(x - mx) : 0.f;
  red[t] = e; __syncthreads();
  for (int o = 128; o > 0; o >>= 1) {
    if (t < o) red[t] += red[t + o];
    __syncthreads();
  }
  const float inv = 1.0f / red[0];
  if (t < L) att[(size_t)b * L + t] = e * inv;
}

// Pooled sum: out[b, d] = sum_l att[b,l] * X[b,l,d]. grid(ceil(D/256), B).
__global__ void pool_kernel(float* __restrict__ out, int ldo,
                            const float* __restrict__ att,
                            const float* __restrict__ X, int ldx,
                            int L, int D) {
  const int b = blockIdx.y;
  const int d = blockIdx.x * 256 + threadIdx.x;
  if (d >= D) return;
  float s = 0.f;
  for (int l = 0; l < L; ++l)
    s += att[(size_t)b * L + l] * X[((size_t)b * L + l) * ldx + d];
  out[(size_t)b * ldo + d] = s;
}

// out[b*rows+r, d] = v[...] * q[b, d]
__global__ void bmul_kernel(float* __restrict__ out, const float* __restrict__ v,
                            const float* __restrict__ q, int rowsPerB, int D,
                            long total) {
  const long i = (long)blockIdx.x * 256 + threadIdx.x;
  if (i >= total) return;
  const int d = (int)(i % D);
  const long r = i / D;
  const int b = (int)(r / rowsPerB);
  out[i] = v[i] * q[(size_t)b * D + d];
}

__global__ void add_kernel(float* __restrict__ out, const float* __restrict__ a,
                           const float* __restrict__ b, long n) {
  const long i = (long)blockIdx.x * 256 + threadIdx.x;
  if (i < n) out[i] = a[i] + b[i];
}

// Per-batch entropy/recon contributions (L == 256). One block per batch.
__global__ void loss_b_kernel(const float* __restrict__ attw,
                              const float* __restrict__ rw,
                              const unsigned char* __restrict__ mask,
                              float* __restrict__ cb) {
  __shared__ float red[256];
  const int b = blockIdx.x, t = threadIdx.x;
  const float p = attw[(size_t)b * 256 + t];
  const float r = rw[(size_t)b * 256 + t];
  const float mk = mask[(size_t)b * 256 + t] ? 1.f : 0.f;
  const float pe = (p == 0.f) ? 1e-9f : p;

  red[t] = mk; __syncthreads();
  for (int o = 128; o > 0; o >>= 1) { if (t < o) red[t] += red[t + o]; __syncthreads(); }
  const float cnt = red[0];
  __syncthreads();
  red[t] = -pe * logf(pe); __syncthreads();
  for (int o = 128; o > 0; o >>= 1) { if (t < o) red[t] += red[t + o]; __syncthreads(); }
  const float eattn = red[0];
  __syncthreads();
  red[t] = (p - r) * (p - r); __syncthreads();
  for (int o = 128; o > 0; o >>= 1) { if (t < o) red[t] += red[t + o]; __syncthreads(); }
  if (t == 0) {
    const float on = 256.f - cnt;
    const float eavg = (256.f / on) * logf(on);
    const float rate = ((eattn - eavg) / eavg > 0.f) ? 1.f : 0.f;
    cb[b] = red[0] * rate;
  }
}

__global__ void loss_sum_kernel(const float* __restrict__ cb,
                                const float* __restrict__ temp,
                                float* __restrict__ out) {
  __shared__ float red[64];
  const int t = threadIdx.x;
  red[t] = cb[t]; __syncthreads();
  for (int o = 32; o > 0; o >>= 1) { if (t < o) red[t] += red[t + o]; __syncthreads(); }
  if (t == 0) out[0] = red[0] * temp[0];
}

// ---------------------------------------------------------------------------
// Host orchestration
// ---------------------------------------------------------------------------
static inline void gemm(hipStream_t s, const float* A, int lda, const float* W,
                        int ldw, const float* bias, float* C, int ldc, int M,
                        int N, int K, bool relu) {
  if ((M & 127) == 0) {
    dim3 grid(N / 128, M / 128);
    gemm_big_kernel<<<grid, 256, 0, s>>>(A, lda, W, ldw, bias, C, ldc, K,
                                         relu ? 1 : 0);
  } else {
    dim3 grid(N / 128, M / 64);
    gemm_kernel<<<grid, 256, 0, s>>>(A, lda, W, ldw, bias, C, ldc, K,
                                     relu ? 1 : 0);
  }
}

static inline void attn(hipStream_t s, const float* Q, const float* K,
                        const float* V, float* O, const unsigned char* mask,
                        int B, int Lq, int Lk, int heads, int D, int ld,
                        float scale) {
  dim3 grid(Lq / 16, heads, B);
  attn_kernel<<<grid, 256, 0, s>>>(Q, K, V, O, mask, Lq, Lk, D, ld, scale);
}

// Input indices: jax.tree_util.tree_flatten of setup_inputs() dict
// (alphabetical key order at every nesting level).
#define IN_IMG        0
#define IN_IMG_MASK   1
#define P_AG_LIN_B1   2
#define P_AG_LIN_B2   3
#define P_AG_LIN_W1   4
#define P_AG_LIN_W2   5
#define P_AG_LQ_B1    6
#define P_AG_LQ_B2    7
#define P_AG_LQ_W1    8
#define P_AG_LQ_W2    9
#define P_AG_LV_B1   10
#define P_AG_LV_B2   11
#define P_AG_LV_W1   12
#define P_AG_LV_W2   13
#define P_AFQ_BM     14
#define P_AFQ_MLP_B1 15
#define P_AFQ_MLP_B2 16
#define P_AFQ_MLP_W1 17
#define P_AFQ_MLP_W2 18
#define P_AFQ_WM     19
#define P_BI_BF      20
#define P_IF_BM      21
#define P_IF_MLP_B1  22
#define P_IF_MLP_B2  23
#define P_IF_MLP_W1  24
#define P_IF_MLP_W2  25
#define P_IF_WM      26
#define P_LF_BM      27
#define P_LF_MLP_B1  28
#define P_LF_MLP_B2  29
#define P_LF_MLP_W1  30
#define P_LF_MLP_W2  31
#define P_LF_WM      32
#define P_BI_WF      33
#define P_DEC_FFN_B1 34
#define P_DEC_FFN_B2 35
#define P_DEC_FFN_W1 36
#define P_DEC_FFN_W2 37
#define P_DEC_G1     38
#define P_DEC_G2     39
#define P_DEC_G3     40
#define P_M1_BK      41
#define P_M1_BO      42
#define P_M1_BQ      43
#define P_M1_BV      44
#define P_M1_WK      45
#define P_M1_WO      46
#define P_M1_WQ      47
#define P_M1_WV      48
#define P_M2_BK      49
#define P_M2_BO      50
#define P_M2_BQ      51
#define P_M2_BV      52
#define P_M2_WK      53
#define P_M2_WO      54
#define P_M2_WQ      55
#define P_M2_WV      56
#define P_DEC_N1     57
#define P_DEC_N2     58
#define P_DEC_N3     59
#define P_ENC_FFN_B1 60
#define P_ENC_FFN_B2 61
#define P_ENC_FFN_W1 62
#define P_ENC_FFN_W2 63
#define P_ENC_G1     64
#define P_ENC_G2     65
#define P_E_BK       66
#define P_E_BO       67
#define P_E_BQ       68
#define P_E_BV       69
#define P_E_WK       70
#define P_E_WO       71
#define P_E_WQ       72
#define P_E_WV       73
#define P_ENC_N1     74
#define P_ENC_N2     75
#define P_FINAL_B    76
#define P_FINAL_W    77
#define P_LINQ_B     78
#define P_LINQ_W     79
#define P_PN_B       80
#define P_PN_G       81
#define P_TEMP       82
#define IN_QUE       83
#define IN_QUE_MASK  84

extern "C" void kernel_launch(void* const* d_in, const int* in_sizes, int n_in,
                              void* d_out, int out_size, void* d_ws,
                              size_t ws_size, hipStream_t stream) {
  (void)in_sizes; (void)n_in; (void)out_size; (void)ws_size;
  const int B = 64, NI = 256, SQ = 128, H = 640, H2 = 1280;
  const int RI = B * NI;   // 16384 img rows
  const int RQ = B * SQ;   // 8192  que rows
  const float sc80 = 0.11180339887498948f;   // 1/sqrt(80)
  const float sc640 = 0.039528470752104741f; // 1/sqrt(640)

  #define FP(i) ((const float*)d_in[(i)])
  const float* img = FP(IN_IMG);
  const float* que = FP(IN_QUE);
  const unsigned char* img_mask = (const unsigned char*)d_in[IN_IMG_MASK];
  const unsigned char* que_mask = (const unsigned char*)d_in[IN_QUE_MASK];
  float* out = (float*)d_out;

  // workspace bump allocator (floats)
  float* ws = (float*)d_ws;
  size_t off = 0;
  auto alloc = [&](size_t n) { float* p = ws + off; off += n; return p; };
  float* W1   = alloc((size_t)RI * H2);  // big FFN hidden / spare SI*H pair
  float* T0   = alloc((size_t)RI * H);
  float* T1   = alloc((size_t)RI * H);
  float* T2   = alloc((size_t)RI * H);
  float* T3   = alloc((size_t)RI * H);
  float* QENC = alloc((size_t)RQ * H);
  float* CC   = alloc((size_t)B * H2);   // [l_feat | i_feat] concat rows
  float* PI   = alloc((size_t)B * H);
  float* PL   = alloc((size_t)B * H);
  float* PQ   = alloc((size_t)B * H);
  float* QF   = alloc((size_t)B * H2);
  float* LQ   = alloc((size_t)B * H);
  float* PROJ = alloc((size_t)B * H);
  float* QAG  = alloc((size_t)B * H);
  float* H64  = alloc((size_t)B * H2);
  float* Q2B  = alloc((size_t)B * H);
  float* LOGI = alloc((size_t)RI);
  float* LOGQ = alloc((size_t)RQ);
  float* ATTW = alloc((size_t)B * NI);
  float* ATTL = alloc((size_t)B * SQ);
  float* ATTQ = alloc((size_t)B * SQ);
  float* RW   = alloc((size_t)B * NI);
  float* CB   = alloc((size_t)B);
  float* FOUT = alloc((size_t)B * H2);

  // ====================== Encoder (que: 8192 x 640) ========================
  gemm(stream, que, H, FP(P_E_WQ), H, FP(P_E_BQ), T0, H, RQ, H, H, false);
  gemm(stream, que, H, FP(P_E_WK), H, FP(P_E_BK), T1, H, RQ, H, H, false);
  gemm(stream, que, H, FP(P_E_WV), H, FP(P_E_BV), T2, H, RQ, H, H, false);
  attn(stream, T0, T1, T2, T3, que_mask, B, SQ, SQ, 8, 80, H, sc80);
  gemm(stream, T3, H, FP(P_E_WO), H, FP(P_E_BO), T0, H, RQ, H, H, false);
  ln_kernel<<<RQ, 256, 0, stream>>>(T1, que, T0, FP(P_ENC_G1), FP(P_ENC_N1), H);
  gemm(stream, T1, H, FP(P_ENC_FFN_W1), H2, FP(P_ENC_FFN_B1), W1, H2, RQ, H2, H, true);
  gemm(stream, W1, H2, FP(P_ENC_FFN_W2), H, FP(P_ENC_FFN_B2), T0, H, RQ, H, H2, false);
  ln_kernel<<<RQ, 256, 0, stream>>>(QENC, T1, T0, FP(P_ENC_G2), FP(P_ENC_N2), H);

  // ====================== Decoder (img: 16384 x 640) =======================
  // m1 self-attention
  gemm(stream, img, H, FP(P_M1_WQ), H, FP(P_M1_BQ), T0, H, RI, H, H, false);
  gemm(stream, img, H, FP(P_M1_WK), H, FP(P_M1_BK), T1, H, RI, H, H, false);
  gemm(stream, img, H, FP(P_M1_WV), H, FP(P_M1_BV), T2, H, RI, H, H, false);
  attn(stream, T0, T1, T2, T3, img_mask, B, NI, NI, 8, 80, H, sc80);
  gemm(stream, T3, H, FP(P_M1_WO), H, FP(P_M1_BO), T0, H, RI, H, H, false);
  ln_kernel<<<RI, 256, 0, stream>>>(T1, img, T0, FP(P_DEC_G1), FP(P_DEC_N1), H);
  // m2 cross-attention: Q from x (T1), K/V from encoded que
  gemm(stream, T1, H, FP(P_M2_WQ), H, FP(P_M2_BQ), T0, H, RI, H, H, false);
  gemm(stream, QENC, H, FP(P_M2_WK), H, FP(P_M2_BK), T2, H, RQ, H, H, false);
  gemm(stream, QENC, H, FP(P_M2_WV), H, FP(P_M2_BV), T3, H, RQ, H, H, false);
  attn(stream, T0, T2, T3, W1, que_mask, B, NI, SQ, 8, 80, H, sc80); // att2 -> W1
  gemm(stream, W1, H, FP(P_M2_WO), H, FP(P_M2_BO), T2, H, RI, H, H, false);
  ln_kernel<<<RI, 256, 0, stream>>>(T3, T1, T2, FP(P_DEC_G2), FP(P_DEC_N2), H);
  // FFN
  gemm(stream, T3, H, FP(P_DEC_FFN_W1), H2, FP(P_DEC_FFN_B1), W1, H2, RI, H2, H, true);
  gemm(stream, W1, H2, FP(P_DEC_FFN_W2), H, FP(P_DEC_FFN_B2), T0, H, RI, H, H2, false);
  ln_kernel<<<RI, 256, 0, stream>>>(T1, T3, T0, FP(P_DEC_G3), FP(P_DEC_N3), H);
  // T1 = decoder output x (B, 256, 640)

  // ============ i_feat = _qkv(x, q, q, que_mask) -> T0 (B,256,640) =========
  attn(stream, T1, QENC, QENC, T0, que_mask, B, NI, SQ, 1, H, H, sc640);
  // _attflat(i_feat, img_mask, bi.l_flat)
  gemm(stream, T0, H, FP(P_LF_MLP_W1), H2, FP(P_LF_MLP_B1), W1, H2, RI, H2, H, true);
  gemv1_kernel<<<RI / 8, 256, 0, stream>>>(W1, H2, FP(P_LF_MLP_W2), FP(P_LF_MLP_B2), LOGI, H2);
  softmax_mask_kernel<<<B, 256, 0, stream>>>(ATTW, LOGI, img_mask, NI);
  {
    dim3 g((H + 255) / 256, B);
    pool_kernel<<<g, 256, 0, stream>>>(PI, H, ATTW, T0, H, NI, H);
  }
  gemm(stream, PI, H, FP(P_LF_WM), H, FP(P_LF_BM), CC + H, H2, B, H, H, false);

  // ============ l_feat = _qkv(q, x, x, img_mask) -> T2 (B,128,640) =========
  attn(stream, QENC, T1, T1, T2, img_mask, B, SQ, NI, 1, H, H, sc640);
  // _attflat(l_feat, que_mask, bi.i_flat)
  gemm(stream, T2, H, FP(P_IF_MLP_W1), H2, FP(P_IF_MLP_B1), W1, H2, RQ, H2, H, true);
  gemv1_kernel<<<RQ / 8, 256, 0, stream>>>(W1, H2, FP(P_IF_MLP_W2), FP(P_IF_MLP_B2), LOGQ, H2);
  softmax_mask_kernel<<<B, 256, 0, stream>>>(ATTL, LOGQ, que_mask, SQ);
  {
    dim3 g((H + 255) / 256, B);
    pool_kernel<<<g, 256, 0, stream>>>(PL, H, ATTL, T2, H, SQ, H);
  }
  gemm(stream, PL, H, FP(P_IF_WM), H, FP(P_IF_BM), CC, H2, B, H, H, false);

  // proj_feat = concat([l_feat, i_feat]) @ wf + bf
  gemm(stream, CC, H2, FP(P_BI_WF), H, FP(P_BI_BF), PROJ, H, B, H, H2, false);

  // ======================= attflat_que(q_enc) ==============================
  gemm(stream, QENC, H, FP(P_AFQ_MLP_W1), H2, FP(P_AFQ_MLP_B1), W1, H2, RQ, H2, H, true);
  gemv1_kernel<<<RQ / 8, 256, 0, stream>>>(W1, H2, FP(P_AFQ_MLP_W2), FP(P_AFQ_MLP_B2), LOGQ, H2);
  softmax_mask_kernel<<<B, 256, 0, stream>>>(ATTQ, LOGQ, que_mask, SQ);
  {
    dim3 g((H + 255) / 256, B);
    pool_kernel<<<g, 256, 0, stream>>>(PQ, H, ATTQ, QENC, H, SQ, H);
  }
  gemm(stream, PQ, H, FP(P_AFQ_WM), H2, FP(P_AFQ_BM), QF, H2, B, H2, H, false);
  gemm(stream, QF, H2, FP(P_LINQ_W), H, FP(P_LINQ_B), LQ, H, B, H, H2, false);
  add_kernel<<<((long)B * H + 255) / 256, 256, 0, stream>>>(QAG, PROJ, LQ, (long)B * H);

  // ======================= _agatt(img, QAG, img_mask) ======================
  gemm(stream, img, H, FP(P_AG_LV_W1), H2, FP(P_AG_LV_B1), W1, H2, RI, H2, H, true);
  gemm(stream, W1, H2, FP(P_AG_LV_W2), H, FP(P_AG_LV_B2), T3, H, RI, H, H2, false);
  gemm(stream, QAG, H, FP(P_AG_LQ_W1), H2, FP(P_AG_LQ_B1), H64, H2, B, H2, H, true);
  gemm(stream, H64, H2, FP(P_AG_LQ_W2), H, FP(P_AG_LQ_B2), Q2B, H, B, H, H2, false);
  {
    long total = (long)RI * H;
    bmul_kernel<<<(total + 255) / 256, 256, 0, stream>>>(T0, T3, Q2B, NI, H, total);
  }
  gemm(stream, T0, H, FP(P_AG_LIN_W1), H2, FP(P_AG_LIN_B1), W1, H2, RI, H2, H, true);
  gemv1_kernel<<<RI / 8, 256, 0, stream>>>(W1, H2, FP(P_AG_LIN_W2), FP(P_AG_LIN_B2), LOGI, H2);
  softmax_mask_kernel<<<B, 256, 0, stream>>>(RW, LOGI, img_mask, NI);

  // ============================ recon loss =================================
  loss_b_kernel<<<B, 256, 0, stream>>>(ATTW, RW, img_mask, CB);
  loss_sum_kernel<<<1, 64, 0, stream>>>(CB, FP(P_TEMP), out + (size_t)B * H2);

  // ===================== final projection + layernorm ======================
  gemm(stream, PROJ, H, FP(P_FINAL_W), H2, FP(P_FINAL_B), FOUT, H2, B, H2, H, false);
  ln_kernel<<<B, 256, 0, stream>>>(out, FOUT, nullptr, FP(P_PN_G), FP(P_PN_B), H2);
  #undef FP
}